// CoxHazardLoss_33595234189713
// MI455X (gfx1250) — compile-verified
//
#include <hip/hip_runtime.h>

typedef __attribute__((ext_vector_type(2))) float v2f;
typedef __attribute__((ext_vector_type(8))) float v8f;

#define EPS_F  1e-07f
#define NB     128
#define NN     512

// One workgroup (512 threads = 16 wave32) per batch.
__global__ __launch_bounds__(NN) void cox_batch_kernel(
    const float* __restrict__ pred,
    const float* __restrict__ target,
    const unsigned char* __restrict__ valid,   // jax bool -> 1 byte
    float* __restrict__ ws_loss,
    float* __restrict__ ws_flag)
{
    const int b = blockIdx.x;
    const int t = threadIdx.x;

    __shared__ float sT[NN];     // target_masked
    __shared__ float sE[NN];     // e_j = valid ? exp(pred - gmax) : 0
    __shared__ float sDen[NN];   // den_i from WMMA
    __shared__ float sRed[NN];   // reduction scratch

    const float p  = pred[b * NN + t];
    const float tg = target[b * NN + t];
    const int   vI = valid[b * NN + t] ? 1 : 0;
    const float Tm = vI ? tg : -1.0f;
    sT[t] = Tm;

    // ---- reduction 1: maxT = max(target_masked) ----
    sRed[t] = Tm;
    __syncthreads();
    for (int s = NN / 2; s > 0; s >>= 1) {
        if (t < s) sRed[t] = fmaxf(sRed[t], sRed[t + s]);
        __syncthreads();
    }
    const float maxT = sRed[0];
    __syncthreads();

    // ---- reduction 2: gmax = max(pred over valid) ----
    sRed[t] = vI ? p : -3.402823466e+38f;
    __syncthreads();
    for (int s = NN / 2; s > 0; s >>= 1) {
        if (t < s) sRed[t] = fmaxf(sRed[t], sRed[t + s]);
        __syncthreads();
    }
    const float gmax = sRed[0];
    __syncthreads();

    // ---- reduction 3: valid count ----
    sRed[t] = (float)vI;
    __syncthreads();
    for (int s = NN / 2; s > 0; s >>= 1) {
        if (t < s) sRed[t] += sRed[t + s];
        __syncthreads();
    }
    const float vcnt = sRed[0];
    __syncthreads();

    if (vcnt < 2.0f) {              // uniform branch: whole block exits
        if (t == 0) { ws_loss[b] = 0.0f; ws_flag[b] = 0.0f; }
        return;
    }

    sE[t] = vI ? __expf(p - gmax) : 0.0f;
    __syncthreads();

    // ---- WMMA phase: den_i = sum_j [T_j >= T_i] * e_j ----
    // f32 16x16x4: A = mask tile (16 rows i x 4 k), B = e_k broadcast over 16 cols.
    {
        const int lane = t & 31;
        const int wave = t >> 5;
        const bool hi  = lane >= 16;
        const int  m   = lane & 15;
        for (int r = 0; r < 2; ++r) {
            const int tile = wave * 2 + r;          // 32 tiles of 16 rows
            const int row  = tile * 16 + m;
            const float Ti = sT[row];
            v8f acc = {};
            for (int k = 0; k < NN; k += 4) {
                // A layout: V0 holds K=0 (lanes 0-15) / K=2 (lanes 16-31); V1: K=1 / K=3.
                // B layout analogous; columns are a broadcast of e_k.
                const int k0 = k + (hi ? 2 : 0);
                const int k1 = k + (hi ? 3 : 1);
                v2f a, bb;
                a.x  = (sT[k0] >= Ti) ? 1.0f : 0.0f;
                a.y  = (sT[k1] >= Ti) ? 1.0f : 0.0f;
                bb.x = sE[k0];
                bb.y = sE[k1];
                acc = __builtin_amdgcn_wmma_f32_16x16x4_f32(
                    /*neg_a=*/false, a, /*neg_b=*/false, bb,
                    /*c_mod=*/(short)0, acc, /*reuse_a=*/false, /*reuse_b=*/false);
            }
            // C/D layout: lane covers col N=lane&15; VGPR q -> row q + (hi?8:0).
            if (m == 0) {
                const int rbase = tile * 16 + (hi ? 8 : 0);
#pragma unroll
                for (int q = 0; q < 8; ++q) sDen[rbase + q] = acc[q];
            }
        }
    }
    __syncthreads();

    // ---- per-row loss (thread t owns row i = t) ----
    const float Ti = sT[t];
    const bool is_elim = (Ti < maxT) && (Ti > 0.0f) && (vI != 0);
    float row_loss = 0.0f;
    if (is_elim) {
        const float den     = sDen[t];
        const float logden  = gmax + __logf(den);   // logsumexp over risk set
        row_loss = -(p - logden);                   // part 1
        const float inv_den = 1.0f / den;
        float part2 = 0.0f;
        for (int j = 0; j < NN; ++j) {
            // invalid j has T_j = -1 < Ti (Ti > 0 here), so (Tj>=Ti) implies valid_j
            const float Tj = sT[j];
            if (Tj >= Ti && j != t) {
                float pr = fminf(sE[j] * inv_den, 1.0f - EPS_F);  // probs clamped
                part2   -= __logf(1.0f - pr + EPS_F);
            }
        }
        row_loss += part2;
    }

    // ---- reduce row losses over the batch ----
    sRed[t] = row_loss;
    __syncthreads();
    for (int s = NN / 2; s > 0; s >>= 1) {
        if (t < s) sRed[t] += sRed[t + s];
        __syncthreads();
    }
    if (t == 0) { ws_loss[b] = sRed[0]; ws_flag[b] = 1.0f; }
}

__global__ __launch_bounds__(NB) void cox_final_kernel(
    const float* __restrict__ ws_loss,
    const float* __restrict__ ws_flag,
    float* __restrict__ out)
{
    __shared__ float sl[NB];
    __shared__ float sf[NB];
    const int t = threadIdx.x;
    sl[t] = ws_loss[t];
    sf[t] = ws_flag[t];
    __syncthreads();
    for (int s = NB / 2; s > 0; s >>= 1) {
        if (t < s) { sl[t] += sl[t + s]; sf[t] += sf[t + s]; }
        __syncthreads();
    }
    if (t == 0) out[0] = sl[0] / fmaxf(sf[0], 1.0f);
}

extern "C" void kernel_launch(void* const* d_in, const int* in_sizes, int n_in,
                              void* d_out, int out_size, void* d_ws, size_t ws_size,
                              hipStream_t stream) {
    const float*         pred   = (const float*)d_in[0];
    const float*         target = (const float*)d_in[1];
    const unsigned char* valid  = (const unsigned char*)d_in[2];

    float* ws_loss = (float*)d_ws;
    float* ws_flag = ws_loss + NB;

    cox_batch_kernel<<<NB, NN, 0, stream>>>(pred, target, valid, ws_loss, ws_flag);
    cox_final_kernel<<<1, NB, 0, stream>>>(ws_loss, ws_flag, (float*)d_out);
}